// arw_FoldingNet_33432025432119
// MI455X (gfx1250) — compile-verified
//
#include <hip/hip_runtime.h>

// ---------------- problem constants (match reference) ----------------
#define BB   8
#define NN   4096
#define BN   (BB*NN)        // 32768
#define KNB  16
#define EDG  (BN*KNB)       // 524288 edges
#define H3   64
#define H2   128            // 2*H3
#define H1   512
#define NSW  24

#define LDSP 80             // row-pair stride in float2 units -> bank-conflict-free b64 reads
#define KMAX 128            // max K across both GEMMs

typedef float v2f __attribute__((ext_vector_type(2)));
typedef float v8f __attribute__((ext_vector_type(8)));

__device__ __forceinline__ float selu_f(float v) {
  const float alpha = 1.6732632423543772f;
  const float scale = 1.0507009873554805f;
  return v > 0.f ? scale * v : scale * alpha * (__expf(v) - 1.f);
}

// order-preserving float<->uint for atomicMax over signed floats
__device__ __forceinline__ unsigned f2ord(float f) {
  unsigned u = __float_as_uint(f);
  return (u & 0x80000000u) ? ~u : (u | 0x80000000u);
}
__device__ __forceinline__ float ord2f(unsigned u) {
  return __uint_as_float((u & 0x80000000u) ? (u & 0x7FFFFFFFu) : ~u);
}

// ---------------- 1) local covariance + encoder-1 (12 -> 64) ----------------
__global__ void k_cov_e1(const float* __restrict__ x,
                         const float* __restrict__ w_e1,
                         const float* __restrict__ b_e1,
                         float* __restrict__ h0) {
  __shared__ float sw[12*H3 + H3];
  for (int i = threadIdx.x; i < 12*H3 + H3; i += blockDim.x)
    sw[i] = (i < 12*H3) ? w_e1[i] : b_e1[i - 12*H3];
  __syncthreads();

  int gid = blockIdx.x * blockDim.x + threadIdx.x;
  if (gid >= BN) return;
  int b = gid / NN, n = gid % NN;
  const float* xb = x + (size_t)b * NN * 3;

  int lo = n - NSW/2, hi = n + NSW/2;           // 25 offsets
  float sx=0.f, sy=0.f, sz=0.f, cnt=0.f;
  for (int w = lo; w <= hi; ++w) {
    if (w < 0 || w >= NN) continue;
    sx += xb[w*3+0]; sy += xb[w*3+1]; sz += xb[w*3+2]; cnt += 1.f;
  }
  float inv = 1.f / cnt;
  float mx = sx*inv, my = sy*inv, mz = sz*inv;
  float c00=0.f,c01=0.f,c02=0.f,c11=0.f,c12=0.f,c22=0.f;
  for (int w = lo; w <= hi; ++w) {
    if (w < 0 || w >= NN) continue;
    float dx = xb[w*3+0]-mx, dy = xb[w*3+1]-my, dz = xb[w*3+2]-mz;
    c00 += dx*dx; c01 += dx*dy; c02 += dx*dz;
    c11 += dy*dy; c12 += dy*dz; c22 += dz*dz;
  }
  const float s = 1.f / (float)(NSW - 1);
  float feat[12];
  feat[0]=xb[n*3+0]; feat[1]=xb[n*3+1]; feat[2]=xb[n*3+2];
  feat[3]=c00*s; feat[4]=c01*s; feat[5]=c02*s;
  feat[6]=c01*s; feat[7]=c11*s; feat[8]=c12*s;
  feat[9]=c02*s; feat[10]=c12*s; feat[11]=c22*s;

  float* out = h0 + (size_t)gid * H3;
  for (int j = 0; j < H3; ++j) {
    float acc = sw[12*H3 + j];
    #pragma unroll
    for (int d = 0; d < 12; ++d) acc += feat[d] * sw[d*H3 + j];
    out[j] = selu_f(acc);
  }
}

// ---------------- small utility kernels ----------------
__global__ void k_fill_f32(float* p, float v, int n) {
  int i = blockIdx.x * blockDim.x + threadIdx.x;
  if (i < n) p[i] = v;
}
__global__ void k_fill_u32(unsigned* p, unsigned v, int n) {
  int i = blockIdx.x * blockDim.x + threadIdx.x;
  if (i < n) p[i] = v;
}
__global__ void k_deg_scatter(const int* __restrict__ dst, float* __restrict__ deg) {
  int e = blockIdx.x * blockDim.x + threadIdx.x;
  if (e >= EDG) return;
  atomicAdd(&deg[dst[e]], 1.f);
}
__global__ void k_rsqrt_inplace(float* p, int n) {
  int i = blockIdx.x * blockDim.x + threadIdx.x;
  if (i < n) p[i] = rsqrtf(p[i]);
}

// ---------------- WMMA fp32 GEMM: [BN x K] @ [K x Nout] ----------------
// Block = 256 threads = 8 waves sharing one 64-wide n-tile. The K x 64
// weight block is staged in LDS packed as row-pairs of float2 so each WMMA
// B fragment is a single aligned ds_load_b64 straight into a VGPR pair
// (no shuffle movs). Row-pair stride 80 float2 => lanes 0-15 hit banks
// 0..31, lanes 16-31 banks 32..63: conflict-free. Each wave owns a 16x64
// tile via V_WMMA_F32_16X16X4_F32, A fragment reused across 4 WMMAs/K-step.
// Epilogue scales rows by dinv[row]; writes g (scatter source) and acc
// (scatter accumulator pre-seeded with the self-loop term g[row]).
__global__ void k_gemm_wmma_scale(const float* __restrict__ A,
                                  const float* __restrict__ Bw,
                                  const float* __restrict__ dinv,
                                  float* __restrict__ g,
                                  float* __restrict__ acc_out,
                                  int K, int Nout, int ntiles64) {
  __shared__ float2 sB2[(KMAX/2) * LDSP];    // 40 KB max

  int nt = blockIdx.x % ntiles64;
  int mb = blockIdx.x / ntiles64;            // 128-row block index
  int n0 = nt * 64;

  // cooperative stage: sB2[rp*LDSP + c] = { B[2rp][n0+c], B[2rp+1][n0+c] }
  int kp = K >> 1;
  for (int i = threadIdx.x; i < kp * 64; i += 256) {
    int rp = i >> 6, c = i & 63;
    const float* bsrc = Bw + (size_t)(2 * rp) * Nout + n0 + c;
    float2 v;
    v.x = bsrc[0];
    v.y = bsrc[Nout];
    sB2[rp * LDSP + c] = v;
  }
  __syncthreads();

  int lane = threadIdx.x & 31;
  int widx = threadIdx.x >> 5;               // wave in block: m sub-tile
  int m0 = (mb * 8 + widx) * 16;

  int mrow = m0 + (lane & 15);
  int kof  = (lane >> 4) << 1;               // lanes 0-15 -> K{0,1}; 16-31 -> K{2,3}
  const float* arow = A + (size_t)mrow * K + kof;
  int cl = lane & 15;
  int rpof = kof >> 1;                       // 0 or 1 row-pair offset

  v8f acc0 = {}, acc1 = {}, acc2 = {}, acc3 = {};
  for (int k = 0; k < K; k += 4) {
    v2f av = *(const v2f*)(arow + k);        // contiguous 8B global load per lane
    const float2* bp = &sB2[((k >> 1) + rpof) * LDSP + cl];
    v2f b0 = *(const v2f*)(bp +  0);         // single ds_load_b64 each
    v2f b1 = *(const v2f*)(bp + 16);
    v2f b2 = *(const v2f*)(bp + 32);
    v2f b3 = *(const v2f*)(bp + 48);
    acc0 = __builtin_amdgcn_wmma_f32_16x16x4_f32(false, av, false, b0, (short)0, acc0, false, false);
    acc1 = __builtin_amdgcn_wmma_f32_16x16x4_f32(false, av, false, b1, (short)0, acc1, false, false);
    acc2 = __builtin_amdgcn_wmma_f32_16x16x4_f32(false, av, false, b2, (short)0, acc2, false, false);
    acc3 = __builtin_amdgcn_wmma_f32_16x16x4_f32(false, av, false, b3, (short)0, acc3, false, false);
  }

  // C/D layout: lanes 0-15 -> rows m0..m0+7, lanes 16-31 -> rows m0+8..m0+15
  int srow = m0 + ((lane >> 4) << 3);
  #pragma unroll
  for (int r = 0; r < 8; ++r) {
    int row = srow + r;
    float dv = dinv[row];
    size_t base = (size_t)row * Nout + n0 + cl;
    float v0 = acc0[r] * dv, v1 = acc1[r] * dv, v2 = acc2[r] * dv, v3 = acc3[r] * dv;
    g[base +  0] = v0;  acc_out[base +  0] = v0;
    g[base + 16] = v1;  acc_out[base + 16] = v1;
    g[base + 32] = v2;  acc_out[base + 32] = v2;
    g[base + 48] = v3;  acc_out[base + 48] = v3;
  }
}

// ---------------- edge scatter: acc[dst] += g[src] (8 features / thread) ----
__global__ void k_edge_scatter(const int* __restrict__ src, const int* __restrict__ dst,
                               const float* __restrict__ g, float* __restrict__ acc,
                               int Nout) {
  const int CH = 8;
  int chunks = Nout / CH;
  long long tid = (long long)blockIdx.x * blockDim.x + threadIdx.x;
  long long total = (long long)EDG * chunks;
  if (tid >= total) return;
  int e  = (int)(tid / chunks);
  int c0 = (int)(tid % chunks) * CH;
  int s = src[e], d = dst[e];
  const float* gp = g + (size_t)s * Nout + c0;
  float* ap = acc + (size_t)d * Nout + c0;
  #pragma unroll
  for (int j = 0; j < CH; ++j) atomicAdd(&ap[j], gp[j]);
}

// ---------------- layer-1 epilogue: h = selu(dinv*acc + b) in place --------
__global__ void k_bias_selu(const float* __restrict__ dinv, const float* __restrict__ bias,
                            float* __restrict__ acc, int Nout) {
  int i = blockIdx.x * blockDim.x + threadIdx.x;
  if (i >= BN * Nout) return;
  int row = i / Nout, colc = i % Nout;
  acc[i] = selu_f(dinv[row] * acc[i] + bias[colc]);
}

// ---------------- layer-2 epilogue fused with per-batch max ----------------
__global__ void k_final_max(const float* __restrict__ dinv, const float* __restrict__ bias,
                            const float* __restrict__ acc, unsigned* __restrict__ latU) {
  const int RB = 256;                        // row-subsample factor
  int tid = blockIdx.x * blockDim.x + threadIdx.x;
  if (tid >= BB * H1 * RB) return;
  int col = tid % H1;
  int r0  = (tid / H1) % RB;
  int b   = tid / (H1 * RB);
  unsigned m = 0u;
  for (int r = r0; r < NN; r += RB) {
    int row = b * NN + r;
    float v = selu_f(dinv[row] * acc[(size_t)row * H1 + col] + bias[col]);
    unsigned e = f2ord(v);
    m = (e > m) ? e : m;
  }
  atomicMax(&latU[b * H1 + col], m);
}

// ------- lat @ w_e2 (+selu) and decoder per-batch constants c1,c2 ----------
__global__ void k_lat_e2(const unsigned* __restrict__ latU,
                         const float* __restrict__ w_e2, const float* __restrict__ b_e2,
                         const float* __restrict__ w_d1, const float* __restrict__ b_d1,
                         const float* __restrict__ w_d2, const float* __restrict__ b_d2,
                         float* __restrict__ c12) {     // BB x 8 (c1[0..2], c2[0..2], pad)
  __shared__ float lat[H1];
  __shared__ float lat2[H1];
  int b = blockIdx.x;
  int t = threadIdx.x;                       // 512 threads
  lat[t] = ord2f(latU[b * H1 + t]);
  __syncthreads();
  float a = b_e2[t];
  for (int k = 0; k < H1; ++k) a += lat[k] * w_e2[k * H1 + t];
  lat2[t] = selu_f(a);
  __syncthreads();
  if (t < 6) {
    int j = t % 3;
    const float* wd = (t < 3) ? w_d1 : w_d2;
    const float* bd = (t < 3) ? b_d1 : b_d2;
    float c = bd[j];
    for (int k = 0; k < H1; ++k) c += lat2[k] * wd[k * 3 + j];
    c12[b * 8 + t] = c;
  }
}

// ---------------- folding decoder (grid coords + tiny MLP) -----------------
__global__ void k_decode(const float* __restrict__ c12,
                         const float* __restrict__ w_d1,
                         const float* __restrict__ w_d2,
                         float* __restrict__ out) {
  int gid = blockIdx.x * blockDim.x + threadIdx.x;
  if (gid >= BN) return;
  int b = gid / NN, n = gid % NN;
  int ix = n / 46, iy = n % 46;              // nx=91, ny=46
  float yx = 1.f + (float)ix * (119.f / 90.f);
  float yy = 1.f + (float)iy * (59.f / 45.f);
  float k0 = selu_f(c12[b*8+0] + yx * w_d1[512*3+0] + yy * w_d1[513*3+0]);
  float k1 = selu_f(c12[b*8+1] + yx * w_d1[512*3+1] + yy * w_d1[513*3+1]);
  float k2 = selu_f(c12[b*8+2] + yx * w_d1[512*3+2] + yy * w_d1[513*3+2]);
  float o0 = selu_f(c12[b*8+3] + k0*w_d2[512*3+0] + k1*w_d2[513*3+0] + k2*w_d2[514*3+0]);
  float o1 = selu_f(c12[b*8+4] + k0*w_d2[512*3+1] + k1*w_d2[513*3+1] + k2*w_d2[514*3+1]);
  float o2 = selu_f(c12[b*8+5] + k0*w_d2[512*3+2] + k1*w_d2[513*3+2] + k2*w_d2[514*3+2]);
  out[(size_t)gid*3+0] = o0;
  out[(size_t)gid*3+1] = o1;
  out[(size_t)gid*3+2] = o2;
}

static inline int cdiv(long long a, long long b) { return (int)((a + b - 1) / b); }

extern "C" void kernel_launch(void* const* d_in, const int* in_sizes, int n_in,
                              void* d_out, int out_size, void* d_ws, size_t ws_size,
                              hipStream_t stream) {
  const float* x    = (const float*)d_in[0];
  const int*   knn  = (const int*)  d_in[1];
  const int*   src  = knn;
  const int*   dst  = knn + EDG;
  const float* w_e1 = (const float*)d_in[2];  const float* b_e1 = (const float*)d_in[3];
  const float* w_g1 = (const float*)d_in[4];  const float* b_g1 = (const float*)d_in[5];
  const float* w_g2 = (const float*)d_in[6];  const float* b_g2 = (const float*)d_in[7];
  const float* w_e2 = (const float*)d_in[8];  const float* b_e2 = (const float*)d_in[9];
  const float* w_d1 = (const float*)d_in[10]; const float* b_d1 = (const float*)d_in[11];
  const float* w_d2 = (const float*)d_in[12]; const float* b_d2 = (const float*)d_in[13];
  float* out = (float*)d_out;

  char* ws = (char*)d_ws;
  float*    dinv = (float*)   (ws);                       // 128 KB
  unsigned* latU = (unsigned*)(ws + (1ull << 18));        // 16 KB
  float*    c12  = (float*)   (ws + (1ull << 18) + (1ull << 15));
  float*    h0   = (float*)   (ws + (1ull   << 20));      // BN x 64   (8 MB)
  float*    g1   = (float*)   (ws + (9ull   << 20));      // BN x 128  (16 MB)
  float*    acc1 = (float*)   (ws + (25ull  << 20));      // BN x 128  (16 MB) -> h1
  float*    g2   = (float*)   (ws + (41ull  << 20));      // BN x 512  (64 MB)
  float*    acc2 = (float*)   (ws + (105ull << 20));      // BN x 512  (64 MB)

  // 1) covariance + encoder-1
  k_cov_e1<<<cdiv(BN, 256), 256, 0, stream>>>(x, w_e1, b_e1, h0);

  // 2) degrees -> dinv (self-loop seeds deg=1)
  k_fill_f32<<<cdiv(BN, 256), 256, 0, stream>>>(dinv, 1.f, BN);
  k_deg_scatter<<<cdiv(EDG, 256), 256, 0, stream>>>(dst, dinv);
  k_rsqrt_inplace<<<cdiv(BN, 256), 256, 0, stream>>>(dinv, BN);

  // 3) GCN layer 1: WMMA GEMM (K=64, Nout=128), scatter, epilogue
  k_gemm_wmma_scale<<<(BN/128) * (H2/64), 256, 0, stream>>>(h0, w_g1, dinv, g1, acc1, H3, H2, H2/64);
  k_edge_scatter<<<cdiv((long long)EDG * (H2/8), 256), 256, 0, stream>>>(src, dst, g1, acc1, H2);
  k_bias_selu<<<cdiv((long long)BN * H2, 256), 256, 0, stream>>>(dinv, b_g1, acc1, H2);

  // 4) GCN layer 2: WMMA GEMM (K=128, Nout=512), scatter, fused max epilogue
  k_gemm_wmma_scale<<<(BN/128) * (H1/64), 256, 0, stream>>>(acc1, w_g2, dinv, g2, acc2, H2, H1, H1/64);
  k_edge_scatter<<<cdiv((long long)EDG * (H1/8), 256), 256, 0, stream>>>(src, dst, g2, acc2, H1);
  k_fill_u32<<<cdiv(BB * H1, 256), 256, 0, stream>>>(latU, 0u, BB * H1);
  k_final_max<<<cdiv((long long)BB * H1 * 256, 256), 256, 0, stream>>>(dinv, b_g2, acc2, latU);

  // 5) latent transform + decoder constants, then folding decoder
  k_lat_e2<<<BB, H1, 0, stream>>>(latU, w_e2, b_e2, w_d1, b_d1, w_d2, b_d2, c12);
  k_decode<<<cdiv(BN, 256), 256, 0, stream>>>(c12, w_d1, w_d2, out);
}